// BoosterModelLoss_47193100648860
// MI455X (gfx1250) — compile-verified
//
#include <hip/hip_runtime.h>
#include <hip/hip_bf16.h>
#include <math.h>

// Problem constants (match the reference)
#define B_    64
#define T_    256
#define VG_   1232
#define S_    32
#define LW_   64
#define VW_   8000
#define NTAGS 8
#define LSEQ  65          // 2*S + 1
#define NEGV  (-1e30f)
#define TAGF  0.1f

typedef __attribute__((ext_vector_type(2))) float v2f;
typedef __attribute__((ext_vector_type(8))) float v8f;

// ---------------------------------------------------------------------------
// Kernel A: fused booster (WMMA f32 16x16x4 GEMM vs one-hot tags) +
//           logsumexp over VG + gather of the 65 CTC label log-probs.
// One wave per 16-row (b,t) tile; 8 waves / block; grid = BT/128 = 128 blocks.
// ---------------------------------------------------------------------------
__global__ __launch_bounds__(256) void booster_lse_kernel(
    const float* __restrict__ scores,   // [B*T, VG]
    const int*   __restrict__ glosses,  // [B, S]
    const int*   __restrict__ tag_ids,  // [VG]
    float*       __restrict__ lp_ext)   // [B*T, LSEQ]
{
    __shared__ int   tag_lds[VG_];
    __shared__ float counts[NTAGS];
    __shared__ float tm[8][16][NTAGS];  // per-wave 16-row x 8-tag means

    const int tid  = threadIdx.x;
    const int lane = tid & 31;
    const int w    = tid >> 5;

    if (tid < NTAGS) counts[tid] = 0.0f;
    __syncthreads();
    for (int v = tid; v < VG_; v += 256) {
        int tg = tag_ids[v];
        tag_lds[v] = tg;
        atomicAdd(&counts[tg], 1.0f);   // integer-valued fp sums: exact & order-independent
    }
    __syncthreads();

    const int rowbase = (blockIdx.x * 8 + w) * 16;
    const int n  = lane & 15;   // A: M index; B/C: N (column/tag) index
    const int kh = lane >> 4;   // selects K pair {0,1} vs {2,3} (A & B layout)

    // ---- Pass 1: tag sums via V_WMMA_F32_16X16X4_F32, K-loop over VG ----
    {
        const float* srow = scores + (size_t)(rowbase + n) * VG_;
        v8f acc = {};
        for (int k = 0; k < VG_; k += 4) {
            const int kk = k + kh * 2;
            __builtin_prefetch(srow + kk + 128, 0, 0);
            v2f a;  a.x = srow[kk];  a.y = srow[kk + 1];
            const int t0 = tag_lds[kk], t1 = tag_lds[kk + 1];
            v2f bb; bb.x = (t0 == n) ? 1.0f : 0.0f;
                    bb.y = (t1 == n) ? 1.0f : 0.0f;
            acc = __builtin_amdgcn_wmma_f32_16x16x4_f32(
                    /*neg_a=*/false, a, /*neg_b=*/false, bb,
                    /*c_mod=*/(short)0, acc, /*reuse_a=*/false, /*reuse_b=*/false);
        }
        // C/D layout: VGPR g holds row M = g + 8*kh, column N = lane&15
        if (n < NTAGS) {
            const float inv = 1.0f / counts[n];
            #pragma unroll
            for (int g = 0; g < 8; ++g)
                tm[w][g + 8 * kh][n] = acc[g] * inv;
        }
    }
    __syncthreads();

    // ---- Pass 2: per-row logsumexp of boosted scores + gather 65 labels ----
    for (int r = 0; r < 16; ++r) {
        const int    row  = rowbase + r;
        const float* srow = scores + (size_t)row * VG_;   // L2-resident re-read
        const float* tmr  = tm[w][r];

        float mx = NEGV;
        for (int v = lane; v < VG_; v += 32)
            mx = fmaxf(mx, srow[v] + TAGF * tmr[tag_lds[v]]);
        #pragma unroll
        for (int off = 16; off > 0; off >>= 1)
            mx = fmaxf(mx, __shfl_xor(mx, off, 32));

        float se = 0.0f;
        for (int v = lane; v < VG_; v += 32)
            se += expf(srow[v] + TAGF * tmr[tag_lds[v]] - mx);
        #pragma unroll
        for (int off = 16; off > 0; off >>= 1)
            se += __shfl_xor(se, off, 32);

        const float lse = mx + logf(se);
        const int   b   = row / T_;
        float* outp = lp_ext + (size_t)row * LSEQ;
        for (int l = lane; l < LSEQ; l += 32) {
            const int v = (l & 1) ? glosses[b * S_ + (l >> 1)] : 0;  // BLANK=0
            outp[l] = srow[v] + TAGF * tmr[tag_lds[v]] - lse;
        }
    }
}

// ---------------------------------------------------------------------------
// Kernel B: CTC forward recurrence, one block per batch element.
// ---------------------------------------------------------------------------
__global__ __launch_bounds__(128) void ctc_kernel(
    const float* __restrict__ lp_ext,       // [B*T, LSEQ]
    const int*   __restrict__ glosses,      // [B, S]
    const int*   __restrict__ frames_len,   // [B]
    const int*   __restrict__ glosses_len,  // [B]
    float*       __restrict__ loss_out)     // [B]
{
    __shared__ float alpha[2][LSEQ];
    __shared__ int   ext[LSEQ];
    __shared__ int   skip[LSEQ];

    const int b   = blockIdx.x;
    const int tid = threadIdx.x;
    const float* lp = lp_ext + (size_t)b * T_ * LSEQ;

    if (tid < LSEQ) ext[tid] = (tid & 1) ? glosses[b * S_ + (tid >> 1)] : 0;
    __syncthreads();
    if (tid < LSEQ) {
        const int p2 = (tid >= 2) ? ext[tid - 2] : 0;
        skip[tid] = (ext[tid] != 0) && (ext[tid] != p2);
        alpha[0][tid] = (tid == 0) ? lp[0] : ((tid == 1) ? lp[1] : NEGV);
    }
    __syncthreads();

    const int Tlen = frames_len[b];
    int cur = 0;
    for (int t = 1; t < T_; ++t) {
        float nv = 0.0f;
        if (tid < LSEQ) {
            const float a0 = alpha[cur][tid];
            const float a1 = (tid >= 1) ? alpha[cur][tid - 1] : NEGV;
            const float a2 = (tid >= 2 && skip[tid]) ? alpha[cur][tid - 2] : NEGV;
            const float m  = fmaxf(a0, fmaxf(a1, a2));
            const float s  = expf(a0 - m) + expf(a1 - m) + expf(a2 - m);
            const float cand = m + logf(s) + lp[(size_t)t * LSEQ + tid];
            nv = (t < Tlen) ? cand : a0;
        }
        __syncthreads();
        if (tid < LSEQ) alpha[cur ^ 1][tid] = nv;
        __syncthreads();
        cur ^= 1;
    }

    if (tid == 0) {
        const int idx1 = 2 * glosses_len[b];
        const int idx0 = (idx1 - 1 > 0) ? idx1 - 1 : 0;
        const float a = alpha[cur][idx0];
        const float c = alpha[cur][idx1];
        const float m = fmaxf(a, c);
        float loss = -(m + logf(expf(a - m) + expf(c - m)));
        if (loss > 1e29f) loss = 0.0f;
        loss_out[b] = loss;
    }
}

// ---------------------------------------------------------------------------
// Kernel C: NLL gather-sum (4032 gathers), fixed-order tree reduction.
// ---------------------------------------------------------------------------
__global__ __launch_bounds__(256) void nll_kernel(
    const float* __restrict__ words_output,  // [B, LW, VW]
    const int*   __restrict__ words,         // [B, LW]
    float*       __restrict__ outp)          // [1]
{
    __shared__ float red[256];
    const int tid = threadIdx.x;
    float s = 0.0f;
    const int n = B_ * (LW_ - 1);
    for (int i = tid; i < n; i += 256) {
        const int b = i / (LW_ - 1);
        const int t = i % (LW_ - 1);
        const int tgt = words[b * LW_ + t + 1];
        if (tgt != 0)   // IGNORE_INDEX = 0
            s -= words_output[((size_t)b * LW_ + t) * VW_ + tgt];
    }
    red[tid] = s;
    __syncthreads();
    #pragma unroll
    for (int off = 128; off > 0; off >>= 1) {
        if (tid < off) red[tid] += red[tid + off];
        __syncthreads();
    }
    if (tid == 0) outp[0] = red[0];
}

// ---------------------------------------------------------------------------
// Kernel D: final combine -> (total, recognition_loss, translation_loss)
// ---------------------------------------------------------------------------
__global__ void finalize_kernel(const float* __restrict__ ctc,
                                const float* __restrict__ nll,
                                float* __restrict__ outp)
{
    if (threadIdx.x == 0 && blockIdx.x == 0) {
        float rec = 0.0f;
        for (int i = 0; i < B_; ++i) rec += ctc[i];   // fixed order: deterministic
        const float trans = nll[0];
        outp[0] = rec + trans;   // GLOSS_W = WORD_W = 1.0
        outp[1] = rec;
        outp[2] = trans;
    }
}

extern "C" void kernel_launch(void* const* d_in, const int* in_sizes, int n_in,
                              void* d_out, int out_size, void* d_ws, size_t ws_size,
                              hipStream_t stream)
{
    (void)in_sizes; (void)n_in; (void)out_size; (void)ws_size;
    const int*   glosses     = (const int*)  d_in[0];
    const int*   words       = (const int*)  d_in[1];
    const float* gscores     = (const float*)d_in[2];
    const float* wout        = (const float*)d_in[3];
    const int*   frames_len  = (const int*)  d_in[4];
    const int*   glosses_len = (const int*)  d_in[5];
    const int*   tag_ids     = (const int*)  d_in[6];
    float*       out         = (float*)      d_out;

    // workspace: lp_ext [B*T*LSEQ] | ctc partials [B] | nll partial [1]
    float* lp     = (float*)d_ws;
    float* ctc_ws = lp + (size_t)B_ * T_ * LSEQ;
    float* nll_ws = ctc_ws + B_;

    booster_lse_kernel<<<(B_ * T_) / 128, 256, 0, stream>>>(gscores, glosses, tag_ids, lp);
    ctc_kernel<<<B_, 128, 0, stream>>>(lp, glosses, frames_len, glosses_len, ctc_ws);
    nll_kernel<<<1, 256, 0, stream>>>(wout, words, nll_ws);
    finalize_kernel<<<1, 32, 0, stream>>>(ctc_ws, nll_ws, out);
}